// LoRADense_58995670778338
// MI455X (gfx1250) — compile-verified
//
#include <hip/hip_runtime.h>

typedef __attribute__((ext_vector_type(2))) float v2f;
typedef __attribute__((ext_vector_type(8))) float v8f;
typedef unsigned int u32x4 __attribute__((ext_vector_type(4)));
typedef int i32x4 __attribute__((ext_vector_type(4)));
typedef int i32x8 __attribute__((ext_vector_type(8)));

#define N_TOK 8192
#define DIM   1024
#define FEAT  1024
#define RANK  16

#if defined(__HIP_DEVICE_COMPILE__) && __has_builtin(__builtin_amdgcn_tensor_load_to_lds)
#define HAVE_TDM 1
#else
#define HAVE_TDM 0
#endif
#if __has_include(<hip/amd_detail/amd_gfx1250_TDM.h>)
#define TDM_ARGS6 1   // therock clang-23: 6-arg builtin
#else
#define TDM_ARGS6 0   // ROCm 7.2 clang-22: 5-arg builtin
#endif

// ---------------------------------------------------------------------------
// WMMA helper: D = A(16x4,f32) * B(4x16,f32) + C(16x16,f32), wave32.
// ---------------------------------------------------------------------------
__device__ __forceinline__ v8f wmma_f32_16x16x4(v2f a, v2f b, v8f c) {
#if __has_builtin(__builtin_amdgcn_wmma_f32_16x16x4_f32)
  return __builtin_amdgcn_wmma_f32_16x16x4_f32(
      /*neg_a=*/false, a, /*neg_b=*/false, b,
      /*c_mod=*/(short)0, c, /*reuse_a=*/false, /*reuse_b=*/false);
#else
  asm volatile("v_wmma_f32_16x16x4_f32 %0, %1, %2, %0"
               : "+v"(c)
               : "v"(a), "v"(b));
  return c;
#endif
}

#if HAVE_TDM
// ---------------------------------------------------------------------------
// TDM: DMA a 2D tile (tile_d0 contiguous elems x tile_d1 rows, fp32) from
// global to LDS with row padding. D# packed per CDNA5 ISA ch.8.
//   pad_ivl_enc: pad after 2^(enc+1) DWORDs;  pad_amt_enc: (enc+1) DWORDs pad
// ---------------------------------------------------------------------------
__device__ __forceinline__ void tdm_load_2d_f32(unsigned lds_addr, const float* gptr,
                                                unsigned tile_d0, unsigned tile_d1,
                                                unsigned tensor_d0, unsigned tensor_d1,
                                                unsigned stride0,
                                                unsigned pad_ivl_enc, unsigned pad_amt_enc) {
  unsigned long long ga = (unsigned long long)(uintptr_t)gptr;
  u32x4 g0;
  g0[0] = 0x1u;                                            // count=1, user mode
  g0[1] = lds_addr;                                        // LDS byte offset
  g0[2] = (unsigned)(ga & 0xFFFFFFFFu);                    // global_addr[31:0]
  g0[3] = (unsigned)((ga >> 32) & 0x01FFFFFFu) | (2u << 30); // [56:32] | type=2
  i32x8 g1;
  g1[0] = (int)((2u << 16)                 // data_size = 4B
              | (1u << 20)                 // pad_enable
              | (pad_ivl_enc << 22)
              | (pad_amt_enc << 25));      // wg_mask=0 (no cluster)
  g1[1] = (int)((tensor_d0 & 0xFFFFu) << 16);                       // dim0[15:0] @63:48
  g1[2] = (int)(((tensor_d0 >> 16) & 0xFFFFu) | ((tensor_d1 & 0xFFFFu) << 16));
  g1[3] = (int)(((tensor_d1 >> 16) & 0xFFFFu) | ((tile_d0 & 0xFFFFu) << 16));
  g1[4] = (int)(tile_d1 & 0xFFFFu);                                 // tile_dim2=0
  g1[5] = (int)stride0;                                             // dim0_stride[31:0]
  g1[6] = 0;                                                        // stride hi / dim1_stride lo
  g1[7] = 0;
  i32x4 z4 = {0, 0, 0, 0};
#if TDM_ARGS6
  i32x8 z8 = {0, 0, 0, 0, 0, 0, 0, 0};
  __builtin_amdgcn_tensor_load_to_lds(g0, g1, z4, z4, z8, 0);
#else
  __builtin_amdgcn_tensor_load_to_lds(g0, g1, z4, z4, 0);
#endif
}
#endif  // HAVE_TDM

// ---------------------------------------------------------------------------
// Kernel 1: low_rank[n][r] = sum_d x[n][d] * lora_a[ids[n]][d][r]
// ---------------------------------------------------------------------------
__global__ void lora_lowrank_kernel(const float* __restrict__ x,
                                    const int* __restrict__ ids,
                                    const float* __restrict__ la,
                                    float* __restrict__ lr) {
  int gid = blockIdx.x * blockDim.x + threadIdx.x;   // N_TOK * RANK threads
  int n = gid >> 4;
  int r = gid & 15;
  int aid = ids[n];
  const float* xr = x + (size_t)n * DIM;
  const float* ar = la + (size_t)aid * (DIM * RANK) + r;
  float acc = 0.0f;
#pragma unroll 8
  for (int d = 0; d < DIM; ++d)
    acc = __builtin_fmaf(xr[d], ar[(size_t)d * RANK], acc);
  lr[gid] = acc;
}

// ---------------------------------------------------------------------------
// Kernel 2: fused  out = x @ W + bias + low_rank @ B[ids]
// WG: 256 threads = 8 waves; WG tile 256x128; wave tile 64x64; BK=16.
// Double-buffered LDS; TDM async tile DMA (waves 0/1 issue, all compute);
// one barrier per k-iteration; fp32 WMMA 16x16x4.
// ---------------------------------------------------------------------------
#define BM 256
#define BN 128
#define BK 16
#define KITERS (DIM / BK)   // 64

#if HAVE_TDM
#define XROW 18             // 16 + 2 pad (TDM pad: interval 16 dw, amount 2)
#define WROW 136            // 128 + 8 pad (TDM pad: interval 128 dw, amount 8)
#define XBUF (BM * XROW)    // 4608 floats per buffer
#define WBUF (BK * WROW)    // 2176 floats per buffer
#else
#define XSP 544             // paired-k row stride
#define WSP 136
#endif

__global__ __launch_bounds__(256)
void lora_dense_fused_kernel(const float* __restrict__ x,
                             const float* __restrict__ w,
                             const float* __restrict__ bias,
                             const int* __restrict__ ids,
                             const float* __restrict__ lr,
                             const float* __restrict__ lb,
                             float* __restrict__ out) {
  const int tid  = threadIdx.x;
  const int lane = tid & 31;
  const int wid  = tid >> 5;              // 0..7
  const int wrow = (wid >> 1) * 64;
  const int wcol = (wid & 1) * 64;
  const int half = lane >> 4;
  const int li   = lane & 15;

  const int rowBase = blockIdx.y * BM;
  const int colBase = blockIdx.x * BN;

  v8f acc[4][4] = {};

#if HAVE_TDM
  // =========================== TDM data path ===============================
  __shared__ __align__(16) float xs[2][XBUF];   // row-major [m][k] + pad
  __shared__ __align__(16) float ws[2][WBUF];   // row-major [k][f] + pad

  const unsigned xsBase = (unsigned)(uintptr_t)&xs[0][0];
  const unsigned wsBase = (unsigned)(uintptr_t)&ws[0][0];

  // prologue: DMA tile 0 into buffer 0
  if (wid == 0)
    tdm_load_2d_f32(xsBase, x + (size_t)rowBase * DIM,
                    BK, BM, DIM, N_TOK, DIM, /*ivl 2^4=16*/3, /*amt 2*/1);
  else if (wid == 1)
    tdm_load_2d_f32(wsBase, w + colBase,
                    BN, BK, FEAT, DIM, FEAT, /*ivl 2^7=128*/6, /*amt 8*/7);
  if (wid < 2) __builtin_amdgcn_s_wait_tensorcnt(0);
  __syncthreads();

  for (int it = 0; it < KITERS; ++it) {
    const int cur = it & 1;
    const int nxt = cur ^ 1;

    // ---- async DMA of tile it+1 while computing tile it ----
    if (it + 1 < KITERS) {
      const int k1 = (it + 1) * BK;
      if (wid == 0)
        tdm_load_2d_f32(xsBase + (unsigned)nxt * XBUF * 4,
                        x + (size_t)rowBase * DIM + k1,
                        BK, BM, DIM, N_TOK, DIM, 3, 1);
      else if (wid == 1)
        tdm_load_2d_f32(wsBase + (unsigned)nxt * WBUF * 4,
                        w + (size_t)k1 * FEAT + colBase,
                        BN, BK, FEAT, DIM, FEAT, 6, 7);
    }

    // ---- compute on tile `cur`: 4 substeps x 16 WMMA ----
    const float* xb = xs[cur];
    const float* wb = ws[cur];
#pragma unroll
    for (int kk = 0; kk < BK; kk += 4) {
      v2f af[4], bf[4];
      const int ka = kk + 2 * half;      // A k-pair start (row-major tile)
#pragma unroll
      for (int t = 0; t < 4; ++t) {
        const int m = wrow + t * 16 + li;
        af[t] = *(const v2f*)&xb[m * XROW + ka];           // ds_load_b64
        const int f = wcol + t * 16 + li;
        bf[t].x = wb[ka * WROW + f];                       // ds_load_2addr
        bf[t].y = wb[(ka + 1) * WROW + f];
      }
#pragma unroll
      for (int tm = 0; tm < 4; ++tm)
#pragma unroll
        for (int tn = 0; tn < 4; ++tn)
          acc[tm][tn] = wmma_f32_16x16x4(af[tm], bf[tn], acc[tm][tn]);
    }

    if (wid < 2) __builtin_amdgcn_s_wait_tensorcnt(0);
    __syncthreads();
  }
#else
  // ====================== fallback: vmem + ds staging ======================
  __shared__ __align__(16) float xs[2][8 * XSP];
  __shared__ __align__(16) float ws[2][BK * WSP];

  const int xc = tid & 3;
  const int xm = tid >> 2;
  float4 xv[4], wv[2];

#pragma unroll
  for (int i = 0; i < 4; ++i)
    xv[i] = *(const float4*)&x[(size_t)(rowBase + xm + i * 64) * DIM + xc * 4];
#pragma unroll
  for (int i = 0; i < 2; ++i) {
    int e = tid + i * 256;
    wv[i] = *(const float4*)&w[(size_t)(e >> 5) * FEAT + colBase + (e & 31) * 4];
  }
#pragma unroll
  for (int i = 0; i < 4; ++i) {
    int m2 = (xm + i * 64) * 2;
    *(v2f*)&xs[0][(xc * 2 + 0) * XSP + m2] = (v2f){xv[i].x, xv[i].y};
    *(v2f*)&xs[0][(xc * 2 + 1) * XSP + m2] = (v2f){xv[i].z, xv[i].w};
  }
#pragma unroll
  for (int i = 0; i < 2; ++i) {
    int e = tid + i * 256;
    *(float4*)&ws[0][(e >> 5) * WSP + (e & 31) * 4] = wv[i];
  }
  __syncthreads();

  for (int it = 0; it < KITERS; ++it) {
    const int cur = it & 1;
    const int nxt = cur ^ 1;
    if (it + 1 < KITERS) {
      const int k1 = (it + 1) * BK;
#pragma unroll
      for (int i = 0; i < 4; ++i)
        xv[i] = *(const float4*)&x[(size_t)(rowBase + xm + i * 64) * DIM + k1 + xc * 4];
#pragma unroll
      for (int i = 0; i < 2; ++i) {
        int e = tid + i * 256;
        wv[i] = *(const float4*)&w[(size_t)(k1 + (e >> 5)) * FEAT + colBase + (e & 31) * 4];
      }
    }
    if (it + 2 < KITERS) {
      const int k2 = (it + 2) * BK;
      __builtin_prefetch(&x[(size_t)(rowBase + tid) * DIM + k2], 0, 0);
      if (tid < 64)
        __builtin_prefetch(&w[(size_t)(k2 + (tid >> 2)) * FEAT + colBase + (tid & 3) * 32], 0, 0);
    }
    const float* xb = xs[cur];
    const float* wb = ws[cur];
#pragma unroll
    for (int kk = 0; kk < BK; kk += 4) {
      v2f af[4], bf[4];
      const int kp = (kk >> 1) + half;
      const int kb = kk + 2 * half;
#pragma unroll
      for (int t = 0; t < 4; ++t) {
        const int m = wrow + t * 16 + li;
        af[t] = *(const v2f*)&xb[kp * XSP + m * 2];
        const int f = wcol + t * 16 + li;
        bf[t].x = wb[kb * WSP + f];
        bf[t].y = wb[(kb + 1) * WSP + f];
      }
#pragma unroll
      for (int tm = 0; tm < 4; ++tm)
#pragma unroll
        for (int tn = 0; tn < 4; ++tn)
          acc[tm][tn] = wmma_f32_16x16x4(af[tm], bf[tn], acc[tm][tn]);
    }
    if (it + 1 < KITERS) {
#pragma unroll
      for (int i = 0; i < 4; ++i) {
        int m2 = (xm + i * 64) * 2;
        *(v2f*)&xs[nxt][(xc * 2 + 0) * XSP + m2] = (v2f){xv[i].x, xv[i].y};
        *(v2f*)&xs[nxt][(xc * 2 + 1) * XSP + m2] = (v2f){xv[i].z, xv[i].w};
      }
#pragma unroll
      for (int i = 0; i < 2; ++i) {
        int e = tid + i * 256;
        *(float4*)&ws[nxt][(e >> 5) * WSP + (e & 31) * 4] = wv[i];
      }
    }
    __syncthreads();
  }
#endif  // HAVE_TDM

  // ---- epilogue: bias + per-token rank-16 LoRA delta, then store ----
  // C/D layout: VGPR v -> row (v + half*8), column li.
#pragma unroll
  for (int tm = 0; tm < 4; ++tm) {
#pragma unroll
    for (int v = 0; v < 8; ++v) {
      const int n = rowBase + wrow + tm * 16 + half * 8 + v;
      const int aid = ids[n];
      const float4* lrp = (const float4*)(lr + (size_t)n * RANK);
      const float4 l0 = lrp[0], l1 = lrp[1], l2 = lrp[2], l3 = lrp[3];
      const float* bp = lb + (size_t)aid * (RANK * FEAT);
#pragma unroll
      for (int tn = 0; tn < 4; ++tn) {
        const int f = colBase + wcol + tn * 16 + li;
        float d = bias[f];
        d = __builtin_fmaf(l0.x, bp[0 * FEAT + f], d);
        d = __builtin_fmaf(l0.y, bp[1 * FEAT + f], d);
        d = __builtin_fmaf(l0.z, bp[2 * FEAT + f], d);
        d = __builtin_fmaf(l0.w, bp[3 * FEAT + f], d);
        d = __builtin_fmaf(l1.x, bp[4 * FEAT + f], d);
        d = __builtin_fmaf(l1.y, bp[5 * FEAT + f], d);
        d = __builtin_fmaf(l1.z, bp[6 * FEAT + f], d);
        d = __builtin_fmaf(l1.w, bp[7 * FEAT + f], d);
        d = __builtin_fmaf(l2.x, bp[8 * FEAT + f], d);
        d = __builtin_fmaf(l2.y, bp[9 * FEAT + f], d);
        d = __builtin_fmaf(l2.z, bp[10 * FEAT + f], d);
        d = __builtin_fmaf(l2.w, bp[11 * FEAT + f], d);
        d = __builtin_fmaf(l3.x, bp[12 * FEAT + f], d);
        d = __builtin_fmaf(l3.y, bp[13 * FEAT + f], d);
        d = __builtin_fmaf(l3.z, bp[14 * FEAT + f], d);
        d = __builtin_fmaf(l3.w, bp[15 * FEAT + f], d);
        out[(size_t)n * FEAT + f] = acc[tm][tn][v] + d;
      }
    }
  }
}

// ---------------------------------------------------------------------------
// Launch
// ---------------------------------------------------------------------------
extern "C" void kernel_launch(void* const* d_in, const int* in_sizes, int n_in,
                              void* d_out, int out_size, void* d_ws, size_t ws_size,
                              hipStream_t stream) {
  const float* x    = (const float*)d_in[0];
  const int*   ids  = (const int*)d_in[1];
  const float* w    = (const float*)d_in[2];
  const float* bias = (const float*)d_in[3];
  const float* la   = (const float*)d_in[4];
  const float* lb   = (const float*)d_in[5];
  float* out = (float*)d_out;
  float* lr  = (float*)d_ws;   // N_TOK * RANK floats = 512 KB scratch

  lora_lowrank_kernel<<<(N_TOK * RANK) / 256, 256, 0, stream>>>(x, ids, la, lr);

  dim3 grid(FEAT / BN, N_TOK / BM);
  lora_dense_fused_kernel<<<grid, 256, 0, stream>>>(x, w, bias, ids, lr, lb, out);
}